// SpatialConvolutionLocal_86749749445042
// MI455X (gfx1250) — compile-verified
//
#include <hip/hip_runtime.h>
#include <stdint.h>

typedef float v2f __attribute__((ext_vector_type(2)));
typedef float v8f __attribute__((ext_vector_type(8)));

// Problem constants (match reference)
constexpr int Bn   = 32;    // batch
constexpr int Cin  = 32;
constexpr int IH   = 64, IW = 64;
constexpr int Cout = 64;
constexpr int OHW  = 64;          // OH == OW == 64
constexpr int Kdim = Cin * 9;     // 288 reduction length (c,i,j)
constexpr int KC   = 48;          // K-chunk staged per double-buffer slot
constexpr int NCH  = Kdim / KC;   // 6 chunks
// Padded LDS row strides (floats): even (8B-aligned v2f), 16B-aligned rows for
// b128 writes, conflict-free for the WMMA operand read pattern.
constexpr int LDA = 292;          // 292 % 64 = 36 = 4*9  -> 16 distinct banks
constexpr int LDW = 52;           // 52  % 64 = 52 = 4*13 -> 16 distinct banks

// gfx1250 async global->LDS copy (VGLOBAL encoding, tracked by ASYNCcnt).
// VDST VGPR carries the LDS byte address, VADDR the 64-bit global address.
__device__ __forceinline__ void async_copy_b128(uint32_t lds_byte, const float* g) {
    asm volatile("global_load_async_to_lds_b128 %0, %1, off"
                 :: "v"(lds_byte), "v"(g)
                 : "memory");
}
__device__ __forceinline__ void wait_async0() {
    asm volatile("s_wait_asynccnt 0x0" ::: "memory");
}

__global__ __launch_bounds__(256)
void lc2d_wmma_kernel(const float* __restrict__ x,
                      const float* __restrict__ wgt,
                      const float* __restrict__ bias,
                      float* __restrict__ out)
{
    __shared__ float sA[Bn * LDA];        // 37,376 B : full 32x288 patch matrix
    __shared__ float sW[2][Cout * LDW];   // 2x13,312 B : double-buffered W K-chunk
    // total LDS = 64,000 B

    const int pos = blockIdx.x;           // h*64 + w
    const int h   = pos >> 6;
    const int w   = pos & 63;
    const int tid = threadIdx.x;

    // ---------------- stage A = im2col patch (zero-padded borders) ----------------
    // A[b][r], r = c*9 + i*3 + j  ->  x[b][c][h+i-1][w+j-1]
    #pragma unroll
    for (int q = 0; q < (Bn * Kdim) / 256; ++q) {   // 36 elements per thread
        int e  = tid + q * 256;
        int b  = e / Kdim;
        int r  = e - b * Kdim;
        int c  = r / 9;
        int ij = r - c * 9;
        int i  = ij / 3;
        int j  = ij - i * 3;
        int ih = h + i - 1;
        int iw = w + j - 1;
        float v = 0.0f;
        if ((unsigned)ih < (unsigned)IH && (unsigned)iw < (unsigned)IW)
            v = x[(((size_t)b * Cin + c) * IH + ih) * IW + iw];
        sA[b * LDA + r] = v;
    }

    // Per-pixel weight block: wgt[pos][o][r], contiguous 64*288 floats.
    const float* wbase = wgt + (size_t)pos * (Cout * Kdim);

    // Async staging of one 64 x 48 chunk: 768 b128 transfers, 3 per thread.
    const uint32_t ldsW0 = (uint32_t)(uintptr_t)&sW[0][0];
    const uint32_t ldsW1 = (uint32_t)(uintptr_t)&sW[1][0];
    auto issueW = [&](int ch, int buf) {
        const float* nb = wbase + ch * KC;
        const uint32_t base = buf ? ldsW1 : ldsW0;
        #pragma unroll
        for (int q = 0; q < 3; ++q) {
            int idx = tid + q * 256;
            int o   = idx / (KC / 4);             // 12 float4 per row
            int kk  = (idx - o * (KC / 4)) * 4;
            async_copy_b128(base + (uint32_t)(o * LDW + kk) * 4u,
                            nb + (size_t)o * Kdim + kk);
        }
    };

    // ---------------- wave tiling ----------------
    const int lane = tid & 31;
    const int nl   = lane & 15;       // N (or M) index within half-wave
    const int hi   = lane >> 4;       // half-wave select
    const int wv   = tid >> 5;        // 8 waves
    const int m0   = (wv & 1) << 4;   // batch-tile base (0 or 16)
    const int n0   = (wv >> 1) << 4;  // cout-tile base (0,16,32,48)

    // Seed accumulator with bias[o][h][w] (depends only on N, same across rows).
    const float bv = bias[(size_t)(n0 + nl) * (OHW * OHW) + pos];
    v8f acc = { bv, bv, bv, bv, bv, bv, bv, bv };

    issueW(0, 0);                     // prologue: chunk 0 -> buf0 (overlaps sA staging)

    // ---------------- main loop: async-stream W chunks, 12 WMMAs per chunk ----------
    for (int ch = 0; ch < NCH; ++ch) {
        const int p = ch & 1;

        wait_async0();                // my chunk-ch transfers landed in sW[p]
        __syncthreads();              // everyone's landed; all reads of sW[p^1] done

        if (ch + 2 < NCH) {           // warm L2 for chunk-after-next
            const float* pf = wbase + (ch + 2) * KC;
            int o  = tid / (KC / 4);
            int kk = (tid - o * (KC / 4)) * 4;
            __builtin_prefetch(pf + (size_t)o * Kdim + kk, 0, 1);
        }
        if (ch + 1 < NCH)             // overlap next chunk's DMA with the WMMAs
            issueW(ch + 1, p ^ 1);

        const int kbase = ch * KC;
        #pragma unroll
        for (int kk = 0; kk < KC; kk += 4) {
            // A operand (16x4 f32): v0/v1 = A[m0+nl][k + 2*hi + {0,1}]
            v2f av  = *(const v2f*)&sA[(m0 + nl) * LDA + kbase + kk + 2 * hi];
            // B operand (4x16 f32): v0/v1 = W[k + 2*hi + {0,1}][n0+nl]
            v2f bvv = *(const v2f*)&sW[p][(n0 + nl) * LDW + kk + 2 * hi];
            acc = __builtin_amdgcn_wmma_f32_16x16x4_f32(
                      /*neg_a=*/false, av, /*neg_b=*/false, bvv,
                      /*c_mod=*/(short)0, acc,
                      /*reuse_a=*/false, /*reuse_b=*/false);
        }
    }

    // ---------------- store C tile: out[b][o][h][w] ----------------
    // C/D layout: VGPR v, lanes 0-15 -> M=v, lanes 16-31 -> M=v+8; N = nl.
    const int o = n0 + nl;
    #pragma unroll
    for (int v = 0; v < 8; ++v) {
        int bb = m0 + v + 8 * hi;
        out[((size_t)(bb * Cout + o) << 12) + pos] = acc[v];
    }
}

extern "C" void kernel_launch(void* const* d_in, const int* in_sizes, int n_in,
                              void* d_out, int out_size, void* d_ws, size_t ws_size,
                              hipStream_t stream) {
    const float* x    = (const float*)d_in[0];
    const float* wgt  = (const float*)d_in[1];
    const float* bias = (const float*)d_in[2];
    float* out        = (float*)d_out;
    (void)in_sizes; (void)n_in; (void)out_size; (void)d_ws; (void)ws_size;

    dim3 grid(OHW * OHW);   // 4096 pixels, one per workgroup
    dim3 block(256);        // 8 wave32
    hipLaunchKernelGGL(lc2d_wmma_kernel, grid, block, 0, stream, x, wgt, bias, out);
}